// SMPLNN_18356690223673
// MI455X (gfx1250) — compile-verified
//
#include <hip/hip_runtime.h>
#include <hip/hip_bf16.h>

typedef __attribute__((ext_vector_type(2))) float v2f;
typedef __attribute__((ext_vector_type(8))) float v8f;
typedef __attribute__((ext_vector_type(4))) unsigned int u32x4;
typedef __attribute__((ext_vector_type(4))) int i32x4;
typedef __attribute__((ext_vector_type(8))) int i32x8;

#define N_PTS    65536
#define N_VERTS  6890
#define N_JOINTS 24
#define NTILES   431                   // ceil(6890/16); 431*16 = 6896 (6 padded cols)
#define BPACK_FLOATS (NTILES * 64)     // 27584 floats = 110336 bytes
#define BONEPACK_FLOATS (6 * 64)       // 6 k-steps of 4, WMMA B layout
#define SB_FLOATS ((NTILES + 1) * 64)  // +1 pad tile so the pipelined prefetch is unconditional

// ---------------- workspace layout (bytes) ----------------
#define WS_PIDX_OFF  0                 // 65536 x int
#define WS_BBUF_OFF  262144            // NN B-matrix, WMMA layout
#define WS_BONE_OFF  372480            // T_fwd B-matrix, WMMA layout

// ---------------- output layout (floats) ----------------
#define OUT_XBAR_OFF 0
#define OUT_RBAR_OFF (N_PTS * 3)
#define OUT_TFWD_OFF (N_PTS * 12)

// =====================================================================
// Kernel 0: pack B matrices into CDNA5 WMMA f32 16x16x4 register layout.
//   B (4x16) per tile: lane L<16  -> vgpr0=B[0][L],    vgpr1=B[1][L]
//                      lane L>=16 -> vgpr0=B[2][L-16], vgpr1=B[3][L-16]
//   NN B-col j = [-2vx, -2vy, -2vz, |v|^2]  (padded cols: [0,0,0,1e30])
// =====================================================================
__global__ __launch_bounds__(256) void pack_kernel(const float* __restrict__ verts,
                                                   const float* __restrict__ bones,
                                                   float* __restrict__ Bbuf,
                                                   float* __restrict__ bonepack) {
    int i = blockIdx.x * blockDim.x + threadIdx.x;
    if (i < BPACK_FLOATS) {
        int t = i >> 6;
        int r = i & 63;
        int L = r >> 1;
        int c = r & 1;
        int n = t * 16 + (L & 15);
        float val;
        if (n >= N_VERTS) {
            val = (L >= 16 && c == 1) ? 1.0e30f : 0.0f;   // |v|^2 slot huge -> never argmin
        } else {
            float vx = verts[n * 3 + 0];
            float vy = verts[n * 3 + 1];
            float vz = verts[n * 3 + 2];
            if (L < 16) val = (c == 0) ? (-2.0f * vx) : (-2.0f * vy);
            else        val = (c == 0) ? (-2.0f * vz) : (vx * vx + vy * vy + vz * vz);
        }
        Bbuf[i] = val;
    } else {
        int j = i - BPACK_FLOATS;
        if (j < BONEPACK_FLOATS) {
            int s = j >> 6;
            int r = j & 63;
            int L = r >> 1;
            int c = r & 1;
            int row = s * 4 + ((L < 16) ? 0 : 2) + c;     // K index 0..23
            int col = L & 15;
            bonepack[j] = bones[row * 16 + col];          // (24,4,4) row-major == 24x16
        }
    }
}

// =====================================================================
// Kernel 1: NN argmin. 32 points/wave, software-pipelined B prefetch,
// TDM bulk-stages the 110 KB B buffer into LDS (one DMA per block).
// =====================================================================
__global__ __launch_bounds__(256) void nn_wmma_kernel(const float* __restrict__ xyz,
                                                      const float* __restrict__ Bbuf,
                                                      int* __restrict__ pidx) {
    extern __shared__ float sB[];      // SB_FLOATS floats = 110592 B of the WGP's 320 KB

    // ---- TDM: one tensor_load_to_lds moves the whole B buffer (wave 0) ----
    if (threadIdx.x == 0) {
        unsigned long long ga = (unsigned long long)(const void*)Bbuf;
        u32x4 g0;
        g0[0] = 1u;                                   // count=1, user descriptor
        g0[1] = 0u;                                   // lds_addr = 0 (start of dyn-LDS alloc)
        g0[2] = (unsigned)(ga & 0xffffffffu);         // global_addr[31:0]
        g0[3] = (unsigned)((ga >> 32) & 0x01ffffffu)  // global_addr[56:32]
              | (2u << 30);                           // type = 2 ("image")
        i32x8 g1;
        g1[0] = 0x20000;                              // data_size = 2 -> 4 bytes
        g1[1] = (int)((unsigned)BPACK_FLOATS << 16);  // tensor_dim0[15:0] in [63:48]
        g1[2] = 0x10000;                              // tensor_dim0[31:16]=0, tensor_dim1=1
        g1[3] = (int)((unsigned)BPACK_FLOATS << 16);  // tile_dim0 in [127:112]
        g1[4] = 1;                                    // tile_dim1 = 1
        g1[5] = BPACK_FLOATS;                         // tensor_dim0_stride[31:0]
        g1[6] = 0;
        g1[7] = 0;
        i32x4 zz = {0, 0, 0, 0};
#if __clang_major__ >= 23
        i32x8 zpad = {};
        __builtin_amdgcn_tensor_load_to_lds(g0, g1, zz, zz, zpad, 0);
#else
        __builtin_amdgcn_tensor_load_to_lds(g0, g1, zz, zz, 0);
#endif
        __builtin_amdgcn_s_wait_tensorcnt(0);
    }
    __syncthreads();

    const int lane = threadIdx.x & 31;
    const int wave = (blockIdx.x * 256 + threadIdx.x) >> 5;
    const int m = wave * 32;           // rows m..m+31
    const int col = lane & 15;

    // A fragments: row = [x,y,z,1]; lane<16 holds K=0,1; lane>=16 holds K=2,3
    v2f aLo, aHi;
    if (lane < 16) {
        const float* p = xyz + (size_t)(m + lane) * 3;
        const float* q = xyz + (size_t)(m + 16 + lane) * 3;
        aLo.x = p[0]; aLo.y = p[1];
        aHi.x = q[0]; aHi.y = q[1];
    } else {
        const float* p = xyz + (size_t)(m + lane - 16) * 3;
        const float* q = xyz + (size_t)(m + lane) * 3;
        aLo.x = p[2]; aLo.y = 1.0f;
        aHi.x = q[2]; aHi.y = 1.0f;
    }

    float bestLo[8], bestHi[8];
    int   idxLo[8],  idxHi[8];
#pragma unroll
    for (int r = 0; r < 8; ++r) {
        bestLo[r] = 3.0e38f; bestHi[r] = 3.0e38f;
        idxLo[r] = 0;        idxHi[r] = 0;
    }

    const v8f zero = {};
    v2f b = *(const v2f*)(sB + lane * 2);            // tile 0
    for (int t = 0; t < NTILES; ++t) {
        v2f bn = *(const v2f*)(sB + (t + 1) * 64 + lane * 2);   // prefetch next tile
        v8f dLo = __builtin_amdgcn_wmma_f32_16x16x4_f32(
            false, aLo, false, b, (short)0, zero, false, false);
        v8f dHi = __builtin_amdgcn_wmma_f32_16x16x4_f32(
            false, aHi, false, b, (short)0, zero, false, false);
        int cand = t * 16 + col;
#pragma unroll
        for (int r = 0; r < 8; ++r) {
            float sL = dLo[r];
            bool ltL = sL < bestLo[r];
            idxLo[r]  = ltL ? cand : idxLo[r];
            bestLo[r] = ltL ? sL : bestLo[r];
            float sH = dHi[r];
            bool ltH = sH < bestHi[r];
            idxHi[r]  = ltH ? cand : idxHi[r];
            bestHi[r] = ltH ? sH : bestHi[r];
        }
        b = bn;
    }

    // branchless argmin reduce over 16 columns within each 16-lane half
#pragma unroll
    for (int mask = 8; mask >= 1; mask >>= 1) {
#pragma unroll
        for (int r = 0; r < 8; ++r) {
            float omL = __shfl_xor(bestLo[r], mask, 32);
            int   oiL = __shfl_xor(idxLo[r], mask, 32);
            bool tL = (omL < bestLo[r]) | ((omL == bestLo[r]) & (oiL < idxLo[r]));
            bestLo[r] = tL ? omL : bestLo[r];
            idxLo[r]  = tL ? oiL : idxLo[r];
            float omH = __shfl_xor(bestHi[r], mask, 32);
            int   oiH = __shfl_xor(idxHi[r], mask, 32);
            bool tH = (omH < bestHi[r]) | ((omH == bestHi[r]) & (oiH < idxHi[r]));
            bestHi[r] = tH ? omH : bestHi[r];
            idxHi[r]  = tH ? oiH : idxHi[r];
        }
    }

    if (lane == 0) {
#pragma unroll
        for (int r = 0; r < 8; ++r) {
            pidx[m + r]      = idxLo[r];   // rows m..m+7
            pidx[m + 16 + r] = idxHi[r];   // rows m+16..m+23
        }
    } else if (lane == 16) {
#pragma unroll
        for (int r = 0; r < 8; ++r) {
            pidx[m + 8 + r]  = idxLo[r];   // rows m+8..m+15
            pidx[m + 24 + r] = idxHi[r];   // rows m+24..m+31
        }
    }
}

// =====================================================================
// Kernel 2: T_fwd = W[p_idx] @ bones — 6 chained V_WMMA_F32_16X16X4_F32.
// D layout matches T_fwd row-major [row*16 + col] directly.
// =====================================================================
__global__ __launch_bounds__(256) void tfwd_wmma_kernel(const float* __restrict__ W,
                                                        const float* __restrict__ bonepack,
                                                        const int* __restrict__ pidx,
                                                        float* __restrict__ Tout) {
    const int lane = threadIdx.x & 31;
    const int wave = (blockIdx.x * 256 + threadIdx.x) >> 5;
    const int m = wave * 16;
    const int col = lane & 15;

    v2f b[6];
#pragma unroll
    for (int s = 0; s < 6; ++s) b[s] = *(const v2f*)(bonepack + s * 64 + lane * 2);

    int p = pidx[m + col];
    const float* wrow = W + (size_t)p * N_JOINTS;
    const int kofs = (lane < 16) ? 0 : 2;
    v2f aM[6];
#pragma unroll
    for (int s = 0; s < 6; ++s) aM[s] = *(const v2f*)(wrow + s * 4 + kofs);  // 8B aligned

    v8f acc = {};
#pragma unroll
    for (int s = 0; s < 6; ++s) {
        acc = __builtin_amdgcn_wmma_f32_16x16x4_f32(
            false, aM[s], false, b[s], (short)0, acc, false, false);
    }

    const int rowbase = m + ((lane >= 16) ? 8 : 0);
#pragma unroll
    for (int r = 0; r < 8; ++r) Tout[(size_t)(rowbase + r) * 16 + col] = acc[r];
}

// =====================================================================
// Kernel 3: per-point finish — x_bar, quaternion->R_hat, rotation_bar.
// =====================================================================
__global__ __launch_bounds__(256) void finalize_kernel(const float* __restrict__ xyz,
                                                       const float* __restrict__ quat,
                                                       const float* __restrict__ Tfwd,
                                                       float* __restrict__ xbar,
                                                       float* __restrict__ rbar) {
    int n = blockIdx.x * blockDim.x + threadIdx.x;
    if (n >= N_PTS) return;

    float T[12];
    const float* Tp = Tfwd + (size_t)n * 16;
#pragma unroll
    for (int i = 0; i < 12; ++i) T[i] = Tp[i];

    float x = xyz[n * 3 + 0], y = xyz[n * 3 + 1], z = xyz[n * 3 + 2];
#pragma unroll
    for (int i = 0; i < 3; ++i)
        xbar[(size_t)n * 3 + i] = T[i * 4 + 0] * x + T[i * 4 + 1] * y + T[i * 4 + 2] * z + T[i * 4 + 3];

    float qr = quat[n * 4 + 0], qx = quat[n * 4 + 1], qy = quat[n * 4 + 2], qz = quat[n * 4 + 3];
    float inv = rsqrtf(qr * qr + qx * qx + qy * qy + qz * qz);
    qr *= inv; qx *= inv; qy *= inv; qz *= inv;

    float R[9];
    R[0] = 1.0f - 2.0f * (qy * qy + qz * qz);
    R[1] = 2.0f * (qx * qy - qr * qz);
    R[2] = 2.0f * (qx * qz + qr * qy);
    R[3] = 2.0f * (qx * qy + qr * qz);
    R[4] = 1.0f - 2.0f * (qx * qx + qz * qz);
    R[5] = 2.0f * (qy * qz - qr * qx);
    R[6] = 2.0f * (qx * qz - qr * qy);
    R[7] = 2.0f * (qy * qz + qr * qx);
    R[8] = 1.0f - 2.0f * (qx * qx + qy * qy);

#pragma unroll
    for (int i = 0; i < 3; ++i)
#pragma unroll
        for (int k = 0; k < 3; ++k)
            rbar[(size_t)n * 9 + i * 3 + k] =
                T[i * 4 + 0] * R[0 + k] + T[i * 4 + 1] * R[3 + k] + T[i * 4 + 2] * R[6 + k];
}

extern "C" void kernel_launch(void* const* d_in, const int* in_sizes, int n_in,
                              void* d_out, int out_size, void* d_ws, size_t ws_size,
                              hipStream_t stream) {
    const float* xyz   = (const float*)d_in[0];   // (65536,3)
    const float* quat  = (const float*)d_in[1];   // (65536,4)
    const float* verts = (const float*)d_in[2];   // (6890,3)
    const float* W     = (const float*)d_in[3];   // (6890,24)
    const float* bones = (const float*)d_in[4];   // (24,4,4)

    float* out  = (float*)d_out;
    float* xbar = out + OUT_XBAR_OFF;
    float* rbar = out + OUT_RBAR_OFF;
    float* Tout = out + OUT_TFWD_OFF;

    char* ws = (char*)d_ws;
    int*   pidx     = (int*)(ws + WS_PIDX_OFF);
    float* Bbuf     = (float*)(ws + WS_BBUF_OFF);
    float* bonepack = (float*)(ws + WS_BONE_OFF);

    // 0) pack B matrices into WMMA layout
    {
        int total = BPACK_FLOATS + BONEPACK_FLOATS;
        int blocks = (total + 255) / 256;
        pack_kernel<<<blocks, 256, 0, stream>>>(verts, bones, Bbuf, bonepack);
    }
    // 1) NN argmin: 2048 waves (32 pts each), 8 waves/block -> 256 blocks
    {
        size_t shmem = SB_FLOATS * sizeof(float);   // 110592 B
        nn_wmma_kernel<<<256, 256, shmem, stream>>>(xyz, Bbuf, pidx);
    }
    // 2) T_fwd GEMM: 4096 waves -> 512 blocks
    tfwd_wmma_kernel<<<512, 256, 0, stream>>>(W, bonepack, pidx, Tout);
    // 3) per-point finish
    finalize_kernel<<<(N_PTS + 255) / 256, 256, 0, stream>>>(xyz, quat, Tout, xbar, rbar);
}